// HouseholderTower_10393820856968
// MI455X (gfx1250) — compile-verified
//
#include <hip/hip_runtime.h>
#include <hip/hip_bf16.h>

typedef __attribute__((ext_vector_type(2))) float v2f;
typedef __attribute__((ext_vector_type(8))) float v8f;

// ---------------------------------------------------------------------------
// Kernel 1: build P = M0*M1*...*M{K-1}, Mk = I - vk vk^T / (vk.vk), and store
// it into d_ws pre-swizzled into V_WMMA_F32_16X16X4_F32 B-fragment order.
//
// B-fragment layout (4x16 f32, 2 VGPRs, wave32):
//   VGPR vi, lane l:  K = 4*s + 2*(l/16) + vi,  N = 16*t + (l%16)
// Swizzled storage (floats): Pswz[ ((t*64 + s)*32 + lane)*2 + vi ]
// One 256-thread block; thread r owns row r of P in registers.
// ---------------------------------------------------------------------------
__global__ __launch_bounds__(256) void hh_build_p(const float* __restrict__ V,
                                                  float* __restrict__ Pswz,
                                                  int K) {
  __shared__ float vsh[256];
  const int r = threadIdx.x; // row index of P, 0..255

  float row[256];
#pragma unroll
  for (int j = 0; j < 256; ++j) row[j] = (j == r) ? 1.0f : 0.0f;

#pragma unroll 1
  for (int k = 0; k < K; ++k) {
    vsh[r] = V[k * 256 + r];
    __syncthreads();

    float vv = 0.0f, hv = 0.0f;
#pragma unroll
    for (int j = 0; j < 256; ++j) {
      const float vj = vsh[j];
      vv = __builtin_fmaf(vj, vj, vv);
      hv = __builtin_fmaf(row[j], vj, hv);
    }
    const float coef = hv / vv;
#pragma unroll
    for (int j = 0; j < 256; ++j) {
      row[j] = __builtin_fmaf(-coef, vsh[j], row[j]);
    }
    __syncthreads();
  }

  // Scatter row r (K-index = r) into WMMA-B fragment order.
  const int s  = r >> 2;         // k-step (K = 4*s + ...)
  const int hi = (r >> 1) & 1;   // lane half (lanes 16-31 carry K+2)
  const int vi = r & 1;          // VGPR index within the v2f fragment
#pragma unroll
  for (int j = 0; j < 256; ++j) {
    const int t    = j >> 4;
    const int lane = hi * 16 + (j & 15);
    Pswz[(((t * 64 + s) * 32 + lane) << 1) + vi] = row[j];
  }
}

// ---------------------------------------------------------------------------
// Kernel 2: Out = H @ P via V_WMMA_F32_16X16X4_F32.
// One wave per 16-row strip; 8 waves (256 threads) per block.
// A-strip (16x256) preloaded into 64 v2f fragments per the ISA A layout:
//   VGPR vi, lane l: M = l%16, K = 4*s + 2*(l/16) + vi
// H is streamed (non-temporal, read once); P fragments hit L2/WGP$;
// Out is streamed with non-temporal stores (written once).
// ---------------------------------------------------------------------------
__global__ __launch_bounds__(256) void hh_gemm(const float* __restrict__ H,
                                               const float* __restrict__ Pswz,
                                               float* __restrict__ Out,
                                               int nStrips) {
  const int lane  = threadIdx.x & 31;
  const int wave  = threadIdx.x >> 5;
  const int strip = blockIdx.x * 8 + wave;
  if (strip >= nStrips) return;

  const int m     = lane & 15;       // row within strip (this lane's M)
  const int khalf = lane >> 4;       // lanes 16-31 carry K+2

  // ---- Preload full 16x256 A strip: 64 fragments x v2f (128 VGPRs) ----
  const float* aPtr = H + (size_t)(strip * 16 + m) * 256 + 2 * khalf;
  v2f a[64];
#pragma unroll
  for (int s = 0; s < 64; ++s) {
    a[s] = __builtin_nontemporal_load((const v2f*)(aPtr + s * 4));
  }

  const float* bBase = Pswz + lane * 2;

#pragma unroll 1
  for (int t = 0; t < 16; ++t) {
    const float* bp = bBase + t * 4096; // (t*64 steps)*32 lanes*2 floats
    v8f acc = {};
#pragma unroll
    for (int s = 0; s < 64; ++s) {
      const v2f b = *(const v2f*)(bp + s * 64);
      // D = A x B + C : v_wmma_f32_16x16x4_f32
      acc = __builtin_amdgcn_wmma_f32_16x16x4_f32(
          /*neg_a=*/false, a[s], /*neg_b=*/false, b,
          /*c_mod=*/(short)0, acc, /*reuse_a=*/false, /*reuse_b=*/false);
    }
    // C/D layout: VGPR v, lane l -> M = v + 8*(l/16), N = l%16
    float* op = Out + (size_t)(strip * 16 + 8 * khalf) * 256 + t * 16 + m;
#pragma unroll
    for (int v = 0; v < 8; ++v) {
      __builtin_nontemporal_store(acc[v], op + (size_t)v * 256);
    }
  }
}

extern "C" void kernel_launch(void* const* d_in, const int* in_sizes, int n_in,
                              void* d_out, int out_size, void* d_ws, size_t ws_size,
                              hipStream_t stream) {
  const float* H = (const float*)d_in[0];   // node_feat [N,256] fp32
  // d_in[1] = adj (unused)
  const float* V = (const float*)d_in[2];   // V [K,256] fp32
  float* Out  = (float*)d_out;              // [N,256] fp32
  float* Pswz = (float*)d_ws;               // 256*256 floats = 256 KB scratch

  const int N = in_sizes[0] / 256;
  const int K = in_sizes[2] / 256;

  hh_build_p<<<1, 256, 0, stream>>>(V, Pswz, K);

  const int nStrips = N / 16;               // N = 100000 -> 6250 strips
  const int blocks  = (nStrips + 7) / 8;    // 8 waves per block
  hh_gemm<<<blocks, 256, 0, stream>>>(H, Pswz, Out, nStrips);
}